// HGSALayer_12403865551355
// MI455X (gfx1250) — compile-verified
//
#include <hip/hip_runtime.h>
#include <hip/hip_bf16.h>
#include <hip/hip_fp16.h>

#define N_NODES   20000
#define N_EDGES   2000
#define IN_FEATS  128
#define NUM_HEADS 4
#define OUT_FEATS 32
#define EDGE_DIM  64
#define HID       128   // NUM_HEADS * OUT_FEATS
#define NEG_SLOPE 0.2f
#define NTILE     4     // N (column) tiles per wave in the projection GEMM

typedef __attribute__((ext_vector_type(2)))  float    v2f;
typedef __attribute__((ext_vector_type(8)))  float    v8f;

// ---------------------------------------------------------------------------
// Init: zero output + edge accumulators, set per-edge running max to -inf.
// ---------------------------------------------------------------------------
__global__ void hgsa_init_kernel(float* __restrict__ out,
                                 float* __restrict__ unnorm,
                                 float* __restrict__ maxv,
                                 float* __restrict__ sumexp) {
    int i = blockIdx.x * blockDim.x + threadIdx.x;
    if (i < N_NODES * HID)      out[i] = 0.0f;
    if (i < N_EDGES * HID)      unnorm[i] = 0.0f;
    if (i < N_EDGES * NUM_HEADS) { maxv[i] = -__builtin_inff(); sumexp[i] = 0.0f; }
}

// ---------------------------------------------------------------------------
// Kernel 1: feat_src = feat @ fc_w   ([20000x128] x [128x128])
// One wave32 per 16x64 output strip: A fragment loaded once per K-step and
// reused across 4 column tiles (register blocking), 4 f32 WMMAs per K-step.
// ---------------------------------------------------------------------------
__global__ __launch_bounds__(32)
void hgsa_proj_kernel(const float* __restrict__ feat,
                      const float* __restrict__ fcw,
                      float* __restrict__ feat_src) {
    const int lane = threadIdx.x;
    const int half = lane >> 4;    // 0: lanes 0-15, 1: lanes 16-31
    const int l16  = lane & 15;
    const int tg   = blockIdx.x;   // 0..1    (group of 4 N tiles = 64 cols)
    const int tm   = blockIdx.y;   // 0..1249 (M tile)
    const int col0 = tg * (16 * NTILE) + l16;

    v8f acc[NTILE] = {};
    // A 16x4 f32: lane holds A[M = lane%16][K = k0 + 2*half + {0,1}] (b64 load)
    const float* arow = feat + (tm * 16 + l16) * IN_FEATS + half * 2;
    // B 4x16 f32: VGPR0 = row (k0 + 2*half), VGPR1 = row (k0 + 2*half + 1)
    const float* brow = fcw + (half * 2) * HID + col0;

    for (int k0 = 0; k0 < IN_FEATS; k0 += 4) {
        v2f a;
        a[0] = arow[k0 + 0];
        a[1] = arow[k0 + 1];
        #pragma unroll
        for (int t = 0; t < NTILE; ++t) {
            v2f b;
            b[0] = brow[k0 * HID + t * 16];
            b[1] = brow[k0 * HID + HID + t * 16];
            acc[t] = __builtin_amdgcn_wmma_f32_16x16x4_f32(
                /*neg_a=*/false, a, /*neg_b=*/false, b,
                /*c_mod=*/(short)0, acc[t], /*reuse_a=*/false, /*reuse_b=*/false);
        }
    }
    // C 16x16 f32: VGPR r -> row (8*half + r), col = lane%16 (+16 per tile)
    float* crow = feat_src + (tm * 16 + half * 8) * HID + col0;
    #pragma unroll
    for (int t = 0; t < NTILE; ++t) {
        #pragma unroll
        for (int r = 0; r < 8; ++r) crow[r * HID + t * 16] = acc[t][r];
    }
}

// ---------------------------------------------------------------------------
// Float atomic max via int punning (sign-aware, IEEE total order on finite).
// ---------------------------------------------------------------------------
__device__ inline void atomicMaxFloat(float* addr, float val) {
    if (val >= 0.0f)
        atomicMax((int*)addr, __float_as_int(val));
    else
        atomicMin((unsigned int*)addr, (unsigned int)__float_as_int(val));
}

// ---------------------------------------------------------------------------
// Kernel 2: per-pair logits e = leaky_relu(<src_feat,attn_src> + <edge,attn_edge>)
//           + per-edge running max (segment_max).
// ---------------------------------------------------------------------------
__global__ __launch_bounds__(128)
void hgsa_logits_kernel(const float* __restrict__ feat_src,
                        const float* __restrict__ edge_feat,
                        const float* __restrict__ attn_src,
                        const float* __restrict__ attn_edge,
                        const int*   __restrict__ src_idx,
                        const int*   __restrict__ edge_idx,
                        float* __restrict__ e_buf,
                        float* __restrict__ maxv,
                        int P) {
    int p = blockIdx.x * blockDim.x + threadIdx.x;
    if (p >= P) return;
    const float* sf = feat_src + (size_t)src_idx[p] * HID;
    const float* ef = edge_feat + (size_t)edge_idx[p] * EDGE_DIM;
    const int eb = edge_idx[p] * NUM_HEADS;
    #pragma unroll
    for (int h = 0; h < NUM_HEADS; ++h) {
        float es = 0.0f;
        for (int d = 0; d < OUT_FEATS; ++d)
            es = fmaf(sf[h * OUT_FEATS + d], attn_src[h * OUT_FEATS + d], es);
        float ee = 0.0f;
        for (int d = 0; d < EDGE_DIM; ++d)
            ee = fmaf(ef[d], attn_edge[h * EDGE_DIM + d], ee);
        float x  = es + ee;
        float ev = x > 0.0f ? x : NEG_SLOPE * x;
        e_buf[p * NUM_HEADS + h] = ev;
        atomicMaxFloat(&maxv[eb + h], ev);
    }
}

// ---------------------------------------------------------------------------
// Kernel 3: one block per pair. exp(e - max) -> atomic sum_exp and
//           atomic unnormalized message accumulation (normalization deferred;
//           softmax scaling is linear so it can be applied per-edge later).
// ---------------------------------------------------------------------------
__global__ __launch_bounds__(128)
void hgsa_accum_kernel(const float* __restrict__ feat_src,
                       const int*   __restrict__ src_idx,
                       const int*   __restrict__ edge_idx,
                       const float* __restrict__ e_buf,
                       const float* __restrict__ maxv,
                       float* __restrict__ sumexp,
                       float* __restrict__ unnorm) {
    const int p = blockIdx.x;
    const int j = threadIdx.x;          // 0..127  (h = j/32, d = j%32)
    const int h = j >> 5;
    const int edge = edge_idx[p];
    const float ev = e_buf[p * NUM_HEADS + h];
    const float m  = maxv[edge * NUM_HEADS + h];
    const float ex = __expf(ev - m);
    if ((j & 31) == 0) atomicAdd(&sumexp[edge * NUM_HEADS + h], ex);
    const float v = feat_src[(size_t)src_idx[p] * HID + j];
    atomicAdd(&unnorm[edge * HID + j], ex * v);
}

// ---------------------------------------------------------------------------
// Kernel 4: per-edge normalization (in place): hef = unnorm / (sum_exp + eps)
// ---------------------------------------------------------------------------
__global__ void hgsa_norm_kernel(float* __restrict__ unnorm,
                                 const float* __restrict__ sumexp) {
    int i = blockIdx.x * blockDim.x + threadIdx.x;
    if (i >= N_EDGES * HID) return;
    int e = i / HID;
    int h = (i % HID) >> 5;
    unnorm[i] = unnorm[i] / (sumexp[e * NUM_HEADS + h] + 1e-9f);
}

// ---------------------------------------------------------------------------
// Kernel 5: sparse dissemination. H is the 0/1 incidence matrix of
// (src_idx, edge_idx), so H @ hef == scatter-add over the pair list.
// Avoids a 160 MB dense read + 10 GFLOP of multiplies-by-zero.
// ---------------------------------------------------------------------------
__global__ __launch_bounds__(128)
void hgsa_dissem_kernel(const float* __restrict__ hef,
                        const int*   __restrict__ src_idx,
                        const int*   __restrict__ edge_idx,
                        float* __restrict__ out) {
    const int p = blockIdx.x;
    const int j = threadIdx.x;
    atomicAdd(&out[(size_t)src_idx[p] * HID + j],
              hef[(size_t)edge_idx[p] * HID + j]);
}

// ---------------------------------------------------------------------------
extern "C" void kernel_launch(void* const* d_in, const int* in_sizes, int n_in,
                              void* d_out, int out_size, void* d_ws, size_t ws_size,
                              hipStream_t stream) {
    const float* feat      = (const float*)d_in[0];
    const float* edge_feat = (const float*)d_in[1];
    // d_in[2] = H (dense incidence) -- intentionally unused; its sparsity
    // pattern is exactly (src_idx, edge_idx).
    const float* fc_w      = (const float*)d_in[3];
    const float* attn_src  = (const float*)d_in[4];
    const float* attn_edge = (const float*)d_in[5];
    const int*   src_idx   = (const int*)d_in[6];
    const int*   edge_idx  = (const int*)d_in[7];
    float*       out       = (float*)d_out;
    const int P = in_sizes[6];

    // Workspace layout (floats): ~15 MB total
    float* ws       = (float*)d_ws;
    float* feat_src = ws;                                   // N_NODES*HID
    float* e_buf    = feat_src + (size_t)N_NODES * HID;     // P*NUM_HEADS
    float* maxv     = e_buf + (size_t)P * NUM_HEADS;        // N_EDGES*NUM_HEADS
    float* sumexp   = maxv + N_EDGES * NUM_HEADS;           // N_EDGES*NUM_HEADS
    float* unnorm   = sumexp + N_EDGES * NUM_HEADS;         // N_EDGES*HID

    hgsa_init_kernel<<<(N_NODES * HID + 255) / 256, 256, 0, stream>>>(
        out, unnorm, maxv, sumexp);

    hgsa_proj_kernel<<<dim3(HID / (16 * NTILE), N_NODES / 16), 32, 0, stream>>>(
        feat, fc_w, feat_src);

    hgsa_logits_kernel<<<(P + 127) / 128, 128, 0, stream>>>(
        feat_src, edge_feat, attn_src, attn_edge, src_idx, edge_idx,
        e_buf, maxv, P);

    hgsa_accum_kernel<<<P, 128, 0, stream>>>(
        feat_src, src_idx, edge_idx, e_buf, maxv, sumexp, unnorm);

    hgsa_norm_kernel<<<(N_EDGES * HID + 255) / 256, 256, 0, stream>>>(
        unnorm, sumexp);

    hgsa_dissem_kernel<<<P, 128, 0, stream>>>(
        unnorm, src_idx, edge_idx, out);
}